// PointPillarScatter_66503273611384
// MI455X (gfx1250) — compile-verified
//
#include <hip/hip_runtime.h>

// PointPillarScatter for MI455X (gfx1250, wave32).
// Zero-FLOP data-movement op: gather formulation + y-quad vectorization turns
// the channel-major scatter into streaming 128-bit non-temporal stores.

#define CCH   64            // channels
#define NXD   432           // x dim
#define NYD   496           // y dim (divisible by 4)
#define NYQ   (NYD / 4)     // 124 y-quads per row
#define PLANE (NXD * NYD)   // 214272 elements per channel plane

typedef float v4f __attribute__((ext_vector_type(4)));
typedef int   v4i __attribute__((ext_vector_type(4)));

// ---------------------------------------------------------------------------
// Kernel 1: map[i] = -1 for i in [0, n). n is a multiple of 4 here; b128 fill.
// ---------------------------------------------------------------------------
__global__ void pp_init_map(int* __restrict__ map, int n) {
    int i = blockIdx.x * blockDim.x + threadIdx.x;   // one int4 per thread
    int base = i * 4;
    if (base + 3 < n) {
        v4i v = { -1, -1, -1, -1 };
        *(v4i*)(map + base) = v;
    } else {
        for (int k = base; k < n; ++k) map[k] = -1;
    }
}

// ---------------------------------------------------------------------------
// Kernel 2: scatter pillar ids into the map. Coords unique per sample
// (reference guarantee) -> deterministic. batch_indices rows are [z, y, x].
// ---------------------------------------------------------------------------
__global__ void pp_scatter_ids(const int* __restrict__ bidx,
                               const int* __restrict__ sidx,
                               int* __restrict__ map, int M) {
    int m = blockIdx.x * blockDim.x + threadIdx.x;
    if (m >= M) return;
    int y = bidx[3 * m + 1];
    int x = bidx[3 * m + 2];
    int b = sidx[m];
    map[b * PLANE + x * NYD + y] = m;
}

// ---------------------------------------------------------------------------
// Kernel 3: gather, one thread per (b, xr, y-quad).
//   grid = (1, NX, B), block = 128 (guard to 124 quads) -> no div/mod at all.
// Per thread:
//   - 1x b128 map load (4 pillar ids, coalesced, L2-hot)
//   - 4x feature rows read via b128 loads (L2-hot, each row read once total)
//   - per channel quad: 4x4 register transpose + 4x NT b128 stores
//     (512 B contiguous per wave per store; output is 219 MB > 192 MB L2 and
//      never re-read, so th:NT keeps map/feats cached while streaming out).
// out[b, c, xr, y] = feats[map[b, NX-1-xr, y], c]   (flip along NX).
// ---------------------------------------------------------------------------
__global__ void pp_gather_out(const float* __restrict__ feats,
                              const int* __restrict__ map,
                              float* __restrict__ out) {
    int q  = blockIdx.x * blockDim.x + threadIdx.x;   // y-quad index
    if (q >= NYQ) return;
    int y  = q * 4;
    int xr = blockIdx.y;                              // output x (flipped)
    int b  = blockIdx.z;

    v4i p = *(const v4i*)(map + (size_t)b * PLANE
                              + (size_t)(NXD - 1 - xr) * NYD + y);

    // Branchless: clamp invalid ids to row 0 (valid memory), scale result by 0.
    int   i0 = (p.x >= 0) ? p.x : 0;  float s0 = (p.x >= 0) ? 1.f : 0.f;
    int   i1 = (p.y >= 0) ? p.y : 0;  float s1 = (p.y >= 0) ? 1.f : 0.f;
    int   i2 = (p.z >= 0) ? p.z : 0;  float s2 = (p.z >= 0) ? 1.f : 0.f;
    int   i3 = (p.w >= 0) ? p.w : 0;  float s3 = (p.w >= 0) ? 1.f : 0.f;

    const v4f* __restrict__ r0 = (const v4f*)(feats + (size_t)i0 * CCH);
    const v4f* __restrict__ r1 = (const v4f*)(feats + (size_t)i1 * CCH);
    const v4f* __restrict__ r2 = (const v4f*)(feats + (size_t)i2 * CCH);
    const v4f* __restrict__ r3 = (const v4f*)(feats + (size_t)i3 * CCH);

    size_t obase = (size_t)b * CCH * PLANE + (size_t)xr * NYD + (size_t)y;

#pragma unroll
    for (int cq = 0; cq < CCH / 4; ++cq) {            // 16 channel quads
        v4f a = r0[cq] * s0;                          // feats[p0, 4c..4c+3]
        v4f bb = r1[cq] * s1;
        v4f c = r2[cq] * s2;
        v4f d = r3[cq] * s3;
        // 4x4 transpose: channel k across the 4 consecutive y positions.
        v4f o0 = { a.x, bb.x, c.x, d.x };
        v4f o1 = { a.y, bb.y, c.y, d.y };
        v4f o2 = { a.z, bb.z, c.z, d.z };
        v4f o3 = { a.w, bb.w, c.w, d.w };
        float* o = out + obase + (size_t)(cq * 4) * PLANE;
        __builtin_nontemporal_store(o0, (v4f*)o);
        __builtin_nontemporal_store(o1, (v4f*)(o + (size_t)PLANE));
        __builtin_nontemporal_store(o2, (v4f*)(o + (size_t)2 * PLANE));
        __builtin_nontemporal_store(o3, (v4f*)(o + (size_t)3 * PLANE));
    }
}

// ---------------------------------------------------------------------------
// Fallback (only if d_ws can't hold the B*NX*NY int map): zero-fill + direct
// scatter. Strictly slower (uncoalesced), kept for robustness.
// ---------------------------------------------------------------------------
__global__ void pp_zero_out(float* __restrict__ out, int n4) {
    int i = blockIdx.x * blockDim.x + threadIdx.x;
    if (i >= n4) return;
    v4f z = { 0.f, 0.f, 0.f, 0.f };
    __builtin_nontemporal_store(z, (v4f*)out + i);    // NT b128 store
}

__global__ void pp_scatter_direct(const float* __restrict__ feats,
                                  const int* __restrict__ bidx,
                                  const int* __restrict__ sidx,
                                  float* __restrict__ out, int M) {
    long long t = (long long)blockIdx.x * blockDim.x + threadIdx.x;
    if (t >= (long long)M * CCH) return;
    int m = (int)(t / CCH);
    int c = (int)(t % CCH);
    int y = bidx[3 * m + 1];
    int x = bidx[3 * m + 2];
    int b = sidx[m];
    size_t o = (size_t)b * CCH * PLANE + (size_t)c * PLANE
             + (size_t)(NXD - 1 - x) * NYD + (size_t)y;
    out[o] = feats[(size_t)m * CCH + c];
}

// ---------------------------------------------------------------------------
extern "C" void kernel_launch(void* const* d_in, const int* in_sizes, int n_in,
                              void* d_out, int out_size, void* d_ws, size_t ws_size,
                              hipStream_t stream) {
    const float* feats = (const float*)d_in[0];   // (M, 64) f32
    const int*   bidx  = (const int*)d_in[1];     // (M, 3)  i32 [z,y,x]
    const int*   sidx  = (const int*)d_in[2];     // (M,)    i32
    float*       out   = (float*)d_out;           // (B, 64, NX, NY) f32

    const int M = in_sizes[2];                    // pillar count
    const int B = out_size / (CCH * PLANE);       // derive batch from out size
    const int n = B * PLANE;                      // map entries

    const size_t mapBytes = (size_t)n * sizeof(int);

    if (ws_size >= mapBytes) {
        int* map = (int*)d_ws;
        int nInit = (n + 3) / 4;                  // one int4 per thread
        pp_init_map<<<(nInit + 255) / 256, 256, 0, stream>>>(map, n);
        pp_scatter_ids<<<(M + 255) / 256, 256, 0, stream>>>(bidx, sidx, map, M);
        dim3 grid(1, NXD, B);                     // 128 threads cover 124 quads
        pp_gather_out<<<grid, 128, 0, stream>>>(feats, map, out);
    } else {
        int n4 = out_size / 4;
        pp_zero_out<<<(n4 + 255) / 256, 256, 0, stream>>>(out, n4);
        long long total = (long long)M * CCH;
        int blocks = (int)((total + 255) / 256);
        pp_scatter_direct<<<blocks, 256, 0, stream>>>(feats, bidx, sidx, out, M);
    }
}